// LIF_6717328851060
// MI455X (gfx1250) — compile-verified
//
#include <hip/hip_runtime.h>
#include <cstdint>

// LIF scan over last axis (T=8) of a (64,128,32,32,8) f32 tensor.
// Memory-bound streaming kernel (512 MB traffic -> ~22 us floor @ 23.3 TB/s):
//  - double-buffered global->LDS async pipeline (CDNA5 ASYNCcnt path)
//  - non-temporal hints on both streams (no reuse; keep 192 MB L2 clean)
//  - register scan, b128 everywhere.

#define TILE  256   // threads per block (8 wave32)
#define ITERS 8     // tiles per block
#define T     8     // timesteps (last axis)

// Native clang vector type: valid for __builtin_nontemporal_* and lowers to
// b128 vector memory ops (HIP's float4 is a struct and is rejected).
typedef float v4f __attribute__((ext_vector_type(4)));

__global__ __launch_bounds__(TILE) void lif_async_kernel(
    const float* __restrict__ X, float* __restrict__ Y, int n_neurons)
{
    // Two 8KB tile buffers: TILE neurons * 8 floats each.
    __shared__ v4f tile[2][TILE * 2];

    const int tid = threadIdx.x;
    const long long base = (long long)blockIdx.x * (TILE * ITERS);

    const unsigned lds_slot0 = (unsigned)(uintptr_t)&tile[0][2 * tid];
    const unsigned lds_slot1 = (unsigned)(uintptr_t)&tile[1][2 * tid];

    // Async-prefetch one neuron's 32B (2 x b128) into this thread's LDS slot.
    // Address is clamped (not branched) so every wave issues exactly 2 async
    // ops per tile -> s_wait_asynccnt accounting is uniform and exact.
    // INST_OFFSET is added to BOTH the LDS and global address for async loads,
    // so offset:16 advances both pointers to the second half of the slot.
    // th:TH_LOAD_NT: input is streamed once -> don't pollute L2.
    auto prefetch = [&](long long neuron, unsigned lds) {
        const long long n = (neuron < (long long)n_neurons)
                                ? neuron : (long long)n_neurons - 1;
        const float* gp = X + n * (long long)T;
        asm volatile(
            "global_load_async_to_lds_b128 %0, %1, off th:TH_LOAD_NT\n\t"
            "global_load_async_to_lds_b128 %0, %1, off offset:16 th:TH_LOAD_NT"
            :: "v"(lds), "v"(gp) : "memory");
    };

    // Pipeline prologue: tile 0 in flight.
    prefetch(base + tid, lds_slot0);

#pragma unroll
    for (int i = 0; i < ITERS; ++i) {
        if (i + 1 < ITERS) {
            // Keep next tile in flight, then wait for current tile only
            // (loads retire in order; <=2 outstanding == tile i complete).
            prefetch(base + (long long)(i + 1) * TILE + tid,
                     ((i + 1) & 1) ? lds_slot1 : lds_slot0);
            asm volatile("s_wait_asynccnt 0x2" ::: "memory");
        } else {
            asm volatile("s_wait_asynccnt 0x0" ::: "memory");
        }

        const long long n = base + (long long)i * TILE + tid;
        if (n < (long long)n_neurons) {
            const v4f a = tile[i & 1][2 * tid];       // ds_load_b128
            const v4f b = tile[i & 1][2 * tid + 1];   // ds_load_b128
            const float x[T] = {a.x, a.y, a.z, a.w, b.x, b.y, b.z, b.w};

            float s[T];
            float mem = 0.0f;
#pragma unroll
            for (int t = 0; t < T; ++t) {
                mem = mem * 0.5f + x[t];          // DECAY = 0.5
                const bool fire = (mem >= 1.0f);  // THRESH = 1.0
                s[t] = fire ? 1.0f : 0.0f;        // hard heaviside
                mem  = fire ? 0.0f : mem;         // reset fired neurons
            }

            const v4f o0 = {s[0], s[1], s[2], s[3]};
            const v4f o1 = {s[4], s[5], s[6], s[7]};
            v4f* op = (v4f*)(Y + n * (long long)T);
            __builtin_nontemporal_store(o0, op);      // b128, th:NT
            __builtin_nontemporal_store(o1, op + 1);  // b128, th:NT
        }
    }
}

extern "C" void kernel_launch(void* const* d_in, const int* in_sizes, int n_in,
                              void* d_out, int out_size, void* d_ws, size_t ws_size,
                              hipStream_t stream)
{
    const float* X = (const float*)d_in[0];
    float* Y = (float*)d_out;

    const long long total     = in_sizes[0];        // 64*128*32*32*8 = 67,108,864
    const int       n_neurons = (int)(total / T);   // 8,388,608 independent scans
    const int       per_block = TILE * ITERS;       // 2048 neurons / block
    const int       blocks    = (n_neurons + per_block - 1) / per_block;  // 4096

    lif_async_kernel<<<blocks, TILE, 0, stream>>>(X, Y, n_neurons);
}